// ZoneEmbedding_49460843381258
// MI455X (gfx1250) — compile-verified
//
#include <hip/hip_runtime.h>
#include <hip/hip_bf16.h>

typedef float v2f __attribute__((ext_vector_type(2)));
typedef float v8f __attribute__((ext_vector_type(8)));

#define B_DIM 128
#define P_DIM 128
#define Z_DIM 1000
#define F_DIM 11
#define ZED_DIM 64
#define TED_DIM 32
#define KPAD 12
#define NROWS (B_DIM * P_DIM)      // 16384
#define OUT_STRIDE 96
#define WAVES_PER_BLOCK 8

// ---------------------------------------------------------------------------
// Prep: build combined weight Wc[12][64] in workspace.
//   rows 0-1   : W_spatial -> cols 0-31, zeros elsewhere
//   rows 2-10  : W_feat    -> cols 32-63, zeros elsewhere
//   row  11    : [b_spatial | b_feat]   (bias folded in via A[:,11] == 1.0)
// ---------------------------------------------------------------------------
__global__ void zone_prep_kernel(const float* __restrict__ Wsp,
                                 const float* __restrict__ bsp,
                                 const float* __restrict__ Wft,
                                 const float* __restrict__ bft,
                                 float* __restrict__ Wc) {
  int i = threadIdx.x + blockIdx.x * blockDim.x;
  if (i >= KPAD * ZED_DIM) return;
  int k = i / ZED_DIM;
  int n = i % ZED_DIM;
  float v = 0.0f;
  if (k < 2) {
    if (n < 32) v = Wsp[k * 32 + n];
  } else if (k < 11) {
    if (n >= 32) v = Wft[(k - 2) * 32 + (n - 32)];
  } else {
    v = (n < 32) ? bsp[n] : bft[n - 32];
  }
  Wc[i] = v;
}

// ---------------------------------------------------------------------------
// Main: one wave32 per 16-row M tile. GEMM (16x12)x(12x64) via three
// V_WMMA_F32_16X16X4_F32 per N-tile, + zone_id_table gather + time embed.
// ---------------------------------------------------------------------------
__global__ __launch_bounds__(WAVES_PER_BLOCK * 32)
void zone_embed_kernel(const float* __restrict__ zf,      // (B,P,Z,F)
                       const int*   __restrict__ ids,     // (B,P)
                       const float* __restrict__ tstep,   // (B,1)
                       const float* __restrict__ Wc,      // (12,64) combined
                       const float* __restrict__ Wtime,   // (1,32)
                       const float* __restrict__ btime,   // (32,)
                       const float* __restrict__ ztab,    // (Z,64)
                       float*       __restrict__ out) {   // (B,P,96)
  const int lane  = threadIdx.x & 31;
  const int wave  = threadIdx.x >> 5;
  const int tile  = blockIdx.x * WAVES_PER_BLOCK + wave;  // 0..1023
  const int m0    = tile * 16;

  const int mcol  = lane & 15;          // M for A frag / N within tile for B,C
  const int khalf = (lane >> 4) * 2;    // K offset: 0 (lanes 0-15) or 2 (16-31)

  // --- A-side gather setup: row features base address -----------------------
  const int    mrow  = m0 + mcol;
  const int    zid   = ids[mrow];
  const size_t fbase = ((size_t)mrow * Z_DIM + (size_t)zid) * F_DIM;

  // --- Preload B fragments: Wc is 12x64, B frag lane l: N=(l&15)+16*nt -----
  v2f bfrag[4][3];
#pragma unroll
  for (int nt = 0; nt < 4; ++nt) {
    const int n = nt * 16 + mcol;
#pragma unroll
    for (int ks = 0; ks < 3; ++ks) {
      const int kb = ks * 4 + khalf;
      v2f b;
      b.x = Wc[kb * ZED_DIM + n];
      b.y = Wc[(kb + 1) * ZED_DIM + n];
      bfrag[nt][ks] = b;
    }
  }

  // --- 3 K-steps x 4 N-tiles of V_WMMA_F32_16X16X4_F32 ---------------------
  v8f acc[4] = {};
#pragma unroll
  for (int ks = 0; ks < 3; ++ks) {
    const int kb = ks * 4 + khalf;            // kb in {0,2,4,6,8,10}
    v2f a;
    a.x = zf[fbase + kb];
    // k==11 is the synthetic bias row (value 1.0); clamp the load so the
    // memory access is unconditional (no EXEC games near WMMA).
    const int  k1     = kb + 1;
    const int  k1clmp = (k1 < F_DIM) ? k1 : (F_DIM - 1);
    const float ld    = zf[fbase + k1clmp];
    a.y = (k1 < F_DIM) ? ld : 1.0f;
#pragma unroll
    for (int nt = 0; nt < 4; ++nt) {
      acc[nt] = __builtin_amdgcn_wmma_f32_16x16x4_f32(
          /*neg_a=*/false, a, /*neg_b=*/false, bfrag[nt][ks],
          /*c_mod=*/(short)0, acc[nt], /*reuse_a=*/false, /*reuse_b=*/false);
    }
  }

  // --- Epilogue: cols 0..63 = acc + zone_id_table[ids] ----------------------
  // C/D layout: VGPR r -> M = r + 8*(lane>>4), N = lane&15 (within tile).
  const int mhi = (lane >> 4) * 8;
#pragma unroll
  for (int r = 0; r < 8; ++r) {
    const int    m  = m0 + mhi + r;
    const int    zr = ids[m];
    const size_t tb = (size_t)zr * ZED_DIM;
    const size_t ob = (size_t)m * OUT_STRIDE;
#pragma unroll
    for (int nt = 0; nt < 4; ++nt) {
      const int n = nt * 16 + mcol;
      out[ob + n] = acc[nt][r] + ztab[tb + n];
    }
  }

  // --- Cols 64..95: time embedding (constant b across a 16-row tile) -------
  const float ts = tstep[m0 >> 7];            // b index = m / 128
  const float tv = ts * Wtime[lane] + btime[lane];
#pragma unroll
  for (int r = 0; r < 16; ++r) {
    out[(size_t)(m0 + r) * OUT_STRIDE + 64 + lane] = tv;
  }
}

// ---------------------------------------------------------------------------
extern "C" void kernel_launch(void* const* d_in, const int* in_sizes, int n_in,
                              void* d_out, int out_size, void* d_ws, size_t ws_size,
                              hipStream_t stream) {
  const float* zone_features = (const float*)d_in[0];   // (B,P,Z,F)
  const int*   zone_ids      = (const int*)d_in[1];     // (B,P)
  const float* time_step     = (const float*)d_in[2];   // (B,1)
  const float* W_spatial     = (const float*)d_in[3];   // (2,32)
  const float* b_spatial     = (const float*)d_in[4];   // (32,)
  const float* W_feat        = (const float*)d_in[5];   // (9,32)
  const float* b_feat        = (const float*)d_in[6];   // (32,)
  const float* W_time        = (const float*)d_in[7];   // (1,32)
  const float* b_time        = (const float*)d_in[8];   // (32,)
  const float* zone_id_table = (const float*)d_in[9];   // (Z,64)
  float*       out           = (float*)d_out;           // (B,P,96)

  float* Wc = (float*)d_ws;                             // 12*64 floats

  zone_prep_kernel<<<1, 256, 0, stream>>>(W_spatial, b_spatial, W_feat, b_feat, Wc);

  const int tiles  = NROWS / 16;                        // 1024
  const int blocks = tiles / WAVES_PER_BLOCK;           // 128
  zone_embed_kernel<<<blocks, WAVES_PER_BLOCK * 32, 0, stream>>>(
      zone_features, zone_ids, time_step, Wc, W_time, b_time, zone_id_table, out);
}